// TripletLossHuman_4604204942001
// MI455X (gfx1250) — compile-verified
//
#include <hip/hip_runtime.h>
#include <hip/hip_bf16.h>
#include <math.h>

#define L_LABELS 4096
#define D_DIM    128

typedef __attribute__((ext_vector_type(2))) float v2f;
typedef __attribute__((ext_vector_type(8))) float v8f;

// ---- workspace layout (bytes) ----
#define OFF_FIRST 0u            // L * 4        = 16 KB
#define OFF_PSUM  16384u        // up to 8192 block partials (32 KB)
#define OFF_PCNT  49152u        // 32 KB
#define OFF_SQF   81920u        // L * 4        = 16 KB
#define OFF_F     131072u       // L * D * 4    = 2 MB
#define OFF_G     4194304ull    // L * L * 4    = 64 MB
#define NEED_G    (OFF_G + (unsigned long long)L_LABELS * L_LABELS * 4ull)

// ---- first[l] = min{ i : targets[i]==l }, init to n ----
__global__ void k_init_first(int* __restrict__ first, int nval) {
  int i = blockIdx.x * blockDim.x + threadIdx.x;
  if (i < L_LABELS) first[i] = nval;
}

__global__ void k_first(const int* __restrict__ targets, int* __restrict__ first, int n) {
  int i = blockIdx.x * blockDim.x + threadIdx.x;
  if (i < n) atomicMin(&first[targets[i]], i);
}

// ---- gather representative rows + squared norms ----
__global__ __launch_bounds__(128) void k_gather(const float* __restrict__ x,
                                                const int* __restrict__ first,
                                                float* __restrict__ F,
                                                float* __restrict__ sqF, int n) {
  __shared__ float red[128];
  const int l = blockIdx.x;
  const int k = threadIdx.x;
  const int idx = first[l];
  float v = 0.f;
  if (idx < n) v = x[(size_t)idx * D_DIM + k];
  F[(size_t)l * D_DIM + k] = v;
  red[k] = v * v;
  __syncthreads();
  for (int s = 64; s > 0; s >>= 1) {
    if (k < s) red[k] += red[k + s];
    __syncthreads();
  }
  if (k == 0) sqF[l] = red[0];
}

// ---- G = F * F^T via V_WMMA_F32_16X16X4_F32 ----
// grid: (L/64, L/128), block 256 (8 waves). Wave w computes rows [128*by+16w, +16)
// x cols [64*bx, +64) as 4 accumulators; A operand reused across the 4 WMMAs.
__global__ __launch_bounds__(256) void k_gemm(const float* __restrict__ F,
                                              float* __restrict__ G) {
  const int lane  = threadIdx.x & 31;
  const int wave  = threadIdx.x >> 5;
  const int lmod  = lane & 15;
  const int lhalf = lane >> 4;                  // 0: K=0,1  1: K=2,3 (ISA 16x4 f32 layout)
  const int rowBase = blockIdx.y * 128 + wave * 16;
  const int colBase = blockIdx.x * 64;

  const float* Ap = F + (size_t)(rowBase + lmod) * D_DIM + 2 * lhalf;
  const float* Bp = F + (size_t)(colBase + lmod) * D_DIM + 2 * lhalf;

  v8f c0 = {}, c1 = {}, c2 = {}, c3 = {};
  #pragma unroll 4
  for (int k = 0; k < D_DIM; k += 4) {
    v2f a  = *(const v2f*)(Ap + k);
    v2f b0 = *(const v2f*)(Bp + k);
    v2f b1 = *(const v2f*)(Bp + 16 * D_DIM + k);
    v2f b2 = *(const v2f*)(Bp + 32 * D_DIM + k);
    v2f b3 = *(const v2f*)(Bp + 48 * D_DIM + k);
    c0 = __builtin_amdgcn_wmma_f32_16x16x4_f32(false, a, false, b0, (short)0, c0, false, false);
    c1 = __builtin_amdgcn_wmma_f32_16x16x4_f32(false, a, false, b1, (short)0, c1, false, false);
    c2 = __builtin_amdgcn_wmma_f32_16x16x4_f32(false, a, false, b2, (short)0, c2, false, false);
    c3 = __builtin_amdgcn_wmma_f32_16x16x4_f32(false, a, false, b3, (short)0, c3, false, false);
  }

  // C/D layout: VGPR r -> (M = r + 8*lhalf, N = lane%16)
  const int n0 = colBase + lmod;
  #pragma unroll
  for (int r = 0; r < 8; ++r) {
    const size_t m = (size_t)(rowBase + r + 8 * lhalf) * L_LABELS;
    G[m + n0 +  0] = c0[r];
    G[m + n0 + 16] = c1[r];
    G[m + n0 + 32] = c2[r];
    G[m + n0 + 48] = c3[r];
  }
}

// ---- per-triplet loss term ----
__device__ __forceinline__ float triplet_term(float sq0, float sq1, float sq2,
                                              float g01, float g02) {
  const float d2ap = fmaxf(sq0 + sq1 - 2.f * g01, 1e-12f);
  const float d2an = fmaxf(sq0 + sq2 - 2.f * g02, 1e-12f);
  const float dap = sqrtf(d2ap);
  const float dan = sqrtf(d2an);
  // p_ap = s_ap/(s_ap+s_an) with s = 1/(d+1)  ==  (dan+1)/(dap+dan+2)
  const float p = (dan + 1.f) / (dap + dan + 2.f);
  return -logf(p + 1e-8f);
}

// ---- triplet kernel using precomputed Gram matrix ----
__global__ __launch_bounds__(256) void k_triplet_G(const int* __restrict__ ua,
                                                   const int* __restrict__ first,
                                                   const float* __restrict__ sqF,
                                                   const float* __restrict__ G,
                                                   float* __restrict__ psum,
                                                   int* __restrict__ pcnt,
                                                   int T, int n) {
  __shared__ float ssum[256];
  __shared__ int   scnt[256];
  const int t = blockIdx.x * 256 + threadIdx.x;
  float term = 0.f; int c = 0;
  if (t < T) {
    const int a0 = ua[3 * t + 0];
    const int a1 = ua[3 * t + 1];
    const int a2 = ua[3 * t + 2];
    if (first[a0] < n && first[a1] < n && first[a2] < n) {
      const float g01 = G[(size_t)a0 * L_LABELS + a1];
      const float g02 = G[(size_t)a0 * L_LABELS + a2];
      term = triplet_term(sqF[a0], sqF[a1], sqF[a2], g01, g02);
      c = 1;
    }
  }
  ssum[threadIdx.x] = term; scnt[threadIdx.x] = c;
  __syncthreads();
  for (int s = 128; s > 0; s >>= 1) {
    if (threadIdx.x < s) {
      ssum[threadIdx.x] += ssum[threadIdx.x + s];
      scnt[threadIdx.x] += scnt[threadIdx.x + s];
    }
    __syncthreads();
  }
  if (threadIdx.x == 0) { psum[blockIdx.x] = ssum[0]; pcnt[blockIdx.x] = scnt[0]; }
}

// ---- fallback: direct gathered dot products (used only if ws too small for G) ----
__global__ __launch_bounds__(256) void k_triplet_direct(const int* __restrict__ ua,
                                                        const int* __restrict__ first,
                                                        const float* __restrict__ x,
                                                        float* __restrict__ psum,
                                                        int* __restrict__ pcnt,
                                                        int T, int n) {
  __shared__ float ssum[256];
  __shared__ int   scnt[256];
  const int t = blockIdx.x * 256 + threadIdx.x;
  float term = 0.f; int c = 0;
  if (t < T) {
    const int a0 = ua[3 * t + 0];
    const int a1 = ua[3 * t + 1];
    const int a2 = ua[3 * t + 2];
    const int i0 = first[a0], i1 = first[a1], i2 = first[a2];
    if (i0 < n && i1 < n && i2 < n) {
      const float4* x0 = (const float4*)(x + (size_t)i0 * D_DIM);
      const float4* x1 = (const float4*)(x + (size_t)i1 * D_DIM);
      const float4* x2 = (const float4*)(x + (size_t)i2 * D_DIM);
      float s0 = 0.f, s1 = 0.f, s2 = 0.f, d01 = 0.f, d02 = 0.f;
      #pragma unroll 8
      for (int k = 0; k < D_DIM / 4; ++k) {
        const float4 a = x0[k], b = x1[k], d = x2[k];
        s0  = fmaf(a.x,a.x, fmaf(a.y,a.y, fmaf(a.z,a.z, fmaf(a.w,a.w, s0))));
        s1  = fmaf(b.x,b.x, fmaf(b.y,b.y, fmaf(b.z,b.z, fmaf(b.w,b.w, s1))));
        s2  = fmaf(d.x,d.x, fmaf(d.y,d.y, fmaf(d.z,d.z, fmaf(d.w,d.w, s2))));
        d01 = fmaf(a.x,b.x, fmaf(a.y,b.y, fmaf(a.z,b.z, fmaf(a.w,b.w, d01))));
        d02 = fmaf(a.x,d.x, fmaf(a.y,d.y, fmaf(a.z,d.z, fmaf(a.w,d.w, d02))));
      }
      term = triplet_term(s0, s1, s2, d01, d02);
      c = 1;
    }
  }
  ssum[threadIdx.x] = term; scnt[threadIdx.x] = c;
  __syncthreads();
  for (int s = 128; s > 0; s >>= 1) {
    if (threadIdx.x < s) {
      ssum[threadIdx.x] += ssum[threadIdx.x + s];
      scnt[threadIdx.x] += scnt[threadIdx.x + s];
    }
    __syncthreads();
  }
  if (threadIdx.x == 0) { psum[blockIdx.x] = ssum[0]; pcnt[blockIdx.x] = scnt[0]; }
}

// ---- deterministic final reduction + loss assembly ----
__global__ __launch_bounds__(256) void k_final(const float* __restrict__ psum,
                                               const int* __restrict__ pcnt,
                                               int nb, float* __restrict__ out) {
  __shared__ float ssum[256];
  __shared__ int   scnt[256];
  float s = 0.f; int c = 0;
  for (int i = threadIdx.x; i < nb; i += 256) { s += psum[i]; c += pcnt[i]; }
  ssum[threadIdx.x] = s; scnt[threadIdx.x] = c;
  __syncthreads();
  for (int st = 128; st > 0; st >>= 1) {
    if (threadIdx.x < st) {
      ssum[threadIdx.x] += ssum[threadIdx.x + st];
      scnt[threadIdx.x] += scnt[threadIdx.x + st];
    }
    __syncthreads();
  }
  if (threadIdx.x == 0) {
    const int cnt = scnt[0];
    // loss_human is exactly MARGIN whenever cnt > 0 (the -0.0 * diff term vanishes)
    out[0] = (cnt > 0) ? (0.3f + ssum[0] / (float)cnt) : 0.f;
  }
}

extern "C" void kernel_launch(void* const* d_in, const int* in_sizes, int n_in,
                              void* d_out, int out_size, void* d_ws, size_t ws_size,
                              hipStream_t stream) {
  const float* inputs  = (const float*)d_in[0];
  const int*   targets = (const int*)d_in[1];
  const int*   ua      = (const int*)d_in[2];
  float* out = (float*)d_out;

  const int n = in_sizes[0] / D_DIM;   // 8192
  const int T = in_sizes[2] / 3;       // 200000

  char* ws = (char*)d_ws;
  int*   first = (int*)(ws + OFF_FIRST);
  float* psum  = (float*)(ws + OFF_PSUM);
  int*   pcnt  = (int*)(ws + OFF_PCNT);
  float* sqF   = (float*)(ws + OFF_SQF);
  float* F     = (float*)(ws + OFF_F);
  float* G     = (float*)(ws + OFF_G);

  const int NB = (T + 255) / 256;      // 782 blocks of triplets

  k_init_first<<<(L_LABELS + 255) / 256, 256, 0, stream>>>(first, n);
  k_first<<<(n + 255) / 256, 256, 0, stream>>>(targets, first, n);

  if (ws_size >= NEED_G) {
    k_gather<<<L_LABELS, 128, 0, stream>>>(inputs, first, F, sqF, n);
    dim3 g(L_LABELS / 64, L_LABELS / 128);
    k_gemm<<<g, 256, 0, stream>>>(F, G);
    k_triplet_G<<<NB, 256, 0, stream>>>(ua, first, sqF, G, psum, pcnt, T, n);
  } else {
    k_triplet_direct<<<NB, 256, 0, stream>>>(ua, first, inputs, psum, pcnt, T, n);
  }
  k_final<<<1, 256, 0, stream>>>(psum, pcnt, NB, out);
}